// LinODECell_43061342109921
// MI455X (gfx1250) — compile-verified
//
#include <hip/hip_runtime.h>

typedef __attribute__((ext_vector_type(2))) float v2f;
typedef __attribute__((ext_vector_type(4))) float v4f;
typedef __attribute__((ext_vector_type(8))) float v8f;

#define DDIM    32
#define LSTRIDE 34          // padded LDS row stride (floats); keeps v2f loads 8B-aligned, spreads banks
#define NTERMS  12          // Taylor terms: ||dt*A|| <~ 0.7 -> 0.7^12/12! ~ 3e-11 < fp32 eps

// One block = 256 threads = 8 wave32's. Each wave owns 16 batch columns.
// Block cooperatively builds M_k = (alpha*kernel)^k / k! in LDS; waves run the
// WMMA GEMM  W = M_k @ X0tile  and accumulate Y += dt^k * W per column.
__global__ __launch_bounds__(256) void linode_expm_kernel(
    const float* __restrict__ dt,
    const float* __restrict__ x0,
    const float* __restrict__ kern,
    const float* __restrict__ alpha_p,
    float* __restrict__ out)
{
    __shared__ __align__(16) float Ash[DDIM * LSTRIDE];        // A = alpha*kernel
    __shared__ __align__(16) float Mbuf[2][DDIM * LSTRIDE];    // ping-pong M_k

    const int tid  = threadIdx.x;
    const int lane = tid & 31;
    const int wave = tid >> 5;
    const int n    = lane & 15;   // batch column within tile (N index, same for all vgprs of a lane)
    const int hi   = lane >> 4;   // lane half
    const int btile = blockIdx.x * 128 + wave * 16;

    const float alpha = alpha_p[0];

    // ---- init A and M_1 = A in LDS (each thread fills 4 contiguous elements) ----
    {
        const int e0  = tid * 4;
        const int row = e0 >> 5;
        const int c0  = e0 & 31;
        #pragma unroll
        for (int i = 0; i < 4; ++i) {
            float a = alpha * kern[e0 + i];
            Ash[row * LSTRIDE + c0 + i]     = a;
            Mbuf[0][row * LSTRIDE + c0 + i] = a;
        }
    }

    // ---- load fixed B-operand fragments: X0 tile (32 x 16), K chunks of 4 ----
    // B layout (4x16, 32-bit): lane L, vgpr j -> [K = 2*hi + j][N = L&15]
    const float* x0col = x0 + (size_t)(btile + n) * DDIM;
    v2f vfrag[8];
    #pragma unroll
    for (int c = 0; c < 8; ++c)
        vfrag[c] = *(const v2f*)(x0col + c * 4 + 2 * hi);

    // ---- Y init = k=0 term = x0 in C/D fragment layout ----
    // D layout (16x16 f32): lane L, vgpr r -> [M = r + 8*hi][N = L&15]
    v8f Y[2];
    #pragma unroll
    for (int mb = 0; mb < 2; ++mb) {
        v4f lo = *(const v4f*)(x0col + mb * 16 + 8 * hi);
        v4f hh = *(const v4f*)(x0col + mb * 16 + 8 * hi + 4);
        Y[mb][0] = lo[0]; Y[mb][1] = lo[1]; Y[mb][2] = lo[2]; Y[mb][3] = lo[3];
        Y[mb][4] = hh[0]; Y[mb][5] = hh[1]; Y[mb][6] = hh[2]; Y[mb][7] = hh[3];
    }

    const float dtn = dt[btile + n];   // per-column dt (per-lane scalar)
    float p = dtn;                     // dt^k, starting at k=1

    __syncthreads();

    for (int k = 1; k <= NTERMS; ++k) {
        const float* __restrict__ M = Mbuf[(k - 1) & 1];

        // ---- W = M_k @ V (two 16-row blocks, K=32 as 8 chunks of 4); Y += p*W ----
        #pragma unroll
        for (int mb = 0; mb < 2; ++mb) {
            // A layout (16x4, 32-bit): lane L, vgpr j -> [M = L&15][K = 2*hi + j]
            const float* arow = M + (mb * 16 + n) * LSTRIDE + 2 * hi;
            v8f w = {0.f, 0.f, 0.f, 0.f, 0.f, 0.f, 0.f, 0.f};
            #pragma unroll
            for (int c = 0; c < 8; ++c) {
                v2f af = *(const v2f*)(arow + c * 4);
                w = __builtin_amdgcn_wmma_f32_16x16x4_f32(
                        false, af, false, vfrag[c], (short)0, w, false, false);
            }
            #pragma unroll
            for (int r = 0; r < 8; ++r)
                Y[mb][r] += p * w[r];
        }
        p *= dtn;

        // ---- M_{k+1} = A @ M_k / (k+1) into the other buffer (4 elems/thread) ----
        if (k < NTERMS) {
            float* __restrict__ Mn = Mbuf[k & 1];
            const int e0  = tid * 4;
            const int row = e0 >> 5;
            const int c0  = e0 & 31;
            const float* arow2 = Ash + row * LSTRIDE;
            float a0 = 0.f, a1 = 0.f, a2 = 0.f, a3 = 0.f;
            #pragma unroll
            for (int l = 0; l < DDIM; ++l) {
                float a = arow2[l];
                const float* mr = M + l * LSTRIDE + c0;
                a0 += a * mr[0]; a1 += a * mr[1]; a2 += a * mr[2]; a3 += a * mr[3];
            }
            const float inv = 1.0f / (float)(k + 1);
            Mn[row * LSTRIDE + c0 + 0] = a0 * inv;
            Mn[row * LSTRIDE + c0 + 1] = a1 * inv;
            Mn[row * LSTRIDE + c0 + 2] = a2 * inv;
            Mn[row * LSTRIDE + c0 + 3] = a3 * inv;
        }
        __syncthreads();
    }

    // ---- store Y (D layout -> row-major out[b][d], two b128 stores per block-row) ----
    float* outcol = out + (size_t)(btile + n) * DDIM;
    #pragma unroll
    for (int mb = 0; mb < 2; ++mb) {
        v4f lo, hh;
        lo[0] = Y[mb][0]; lo[1] = Y[mb][1]; lo[2] = Y[mb][2]; lo[3] = Y[mb][3];
        hh[0] = Y[mb][4]; hh[1] = Y[mb][5]; hh[2] = Y[mb][6]; hh[3] = Y[mb][7];
        *(v4f*)(outcol + mb * 16 + 8 * hi)     = lo;
        *(v4f*)(outcol + mb * 16 + 8 * hi + 4) = hh;
    }
}

extern "C" void kernel_launch(void* const* d_in, const int* in_sizes, int n_in,
                              void* d_out, int out_size, void* d_ws, size_t ws_size,
                              hipStream_t stream) {
    const float* dt    = (const float*)d_in[0];   // (B,)
    const float* x0    = (const float*)d_in[1];   // (B, 32)
    const float* kern  = (const float*)d_in[2];   // (32, 32)
    const float* alpha = (const float*)d_in[3];   // scalar
    float* out = (float*)d_out;                   // (B, 32)

    const int B = in_sizes[0];                    // 32768
    const int blocks = B / 128;                   // 128 batch columns per block (8 waves x 16)
    linode_expm_kernel<<<blocks, 256, 0, stream>>>(dt, x0, kern, alpha, out);
}